// onering_conv_layer_17557826306182
// MI455X (gfx1250) — compile-verified
//
#include <hip/hip_runtime.h>

// ---------------------------------------------------------------------------
// one-ring icosphere conv: out = gather7(x, neigh) @ W^T + b
//   x     : [N, 32]  f32
//   neigh : [7N]     i64
//   W     : [32,224] f32 (row-major)
//   b     : [32]     f32
//   out   : [N, 32]  f32
//
// One wave32 per 16-node tile; D = 16x32 via two V_WMMA_F32_16X16X4_F32
// accumulators, K=224 consumed as 7 neighbors x 8 steps of K=4.
// W is staged once per block into LDS, pre-swizzled into the per-lane
// B-fragment stream: slot[step][frag][lane] (8B each) -> ds_load_b64 with
// lanes hitting banks {2l,2l+1}, i.e. all 64 banks, conflict-free.
// ---------------------------------------------------------------------------

typedef float v2f __attribute__((ext_vector_type(2)));
typedef float v8f __attribute__((ext_vector_type(8)));

#define IN_F    32
#define OUT_F   32
#define K_DIM   224            // 7 * IN_F
#define KSTEPS  56             // K_DIM / 4
#define WSLOTS  (KSTEPS * 2 * 32)   // 3584 float2 = 28 KB

__global__ __launch_bounds__(256)
void onering_conv_wmma_f32(const float* __restrict__ x,
                           const long long* __restrict__ neigh,
                           const float* __restrict__ W,
                           const float* __restrict__ bias,
                           float* __restrict__ out,
                           int nNodes, int nTiles)
{
    __shared__ v2f lwb[WSLOTS];   // [step 0..55][frag 0..1][lane 0..31]

    const int tid  = threadIdx.x;
    const int lane = tid & 31;
    const int wave = tid >> 5;
    const int tile = blockIdx.x * (blockDim.x >> 5) + wave;

    // ---- Stage W into LDS as swizzled B fragments (all threads, pre-barrier)
    // slot s: lane=s&31, frag=(s>>5)&1, step=s>>6 ; step = j*8 + kk/4
#pragma unroll
    for (int it = 0; it < WSLOTS / 256; ++it) {
        const int s     = it * 256 + tid;
        const int slane = s & 31;
        const int frag  = (s >> 5) & 1;
        const int step  = s >> 6;
        const int n     = (slane & 15) + 16 * frag;     // output column
        const int k     = step * 4 + ((slane >> 4) * 2); // K index (ko per lane half)
        lwb[s] = *(const v2f*)(W + (size_t)n * K_DIM + k);
    }
    __syncthreads();

    if (tile < nTiles) {
        const int rowBase = tile * 16;
        const int m  = lane & 15;        // matrix row M (mirrored across halves)
        const int ko = (lane >> 4) * 2;  // K sub-offset: lanes 0-15 K{0,1}, 16-31 K{2,3}

        // Clamp gather row for the final partial tile (stores predicated below;
        // EXEC stays all-1s for the WMMAs).
        int node = rowBase + m;
        if (node >= nNodes) node = nNodes - 1;

        // Preload this row's 7 neighbor indices (i64 -> i32, values < N).
        int nidx[7];
#pragma unroll
        for (int j = 0; j < 7; ++j)
            nidx[j] = (int)neigh[7 * (long long)node + j];

        v8f c0 = {};   // output cols  0..15
        v8f c1 = {};   // output cols 16..31

#pragma unroll
        for (int j = 0; j < 7; ++j) {
            const float* xrow = x + (size_t)nidx[j] * IN_F;
            const v2f*   wst  = &lwb[(j * 8) * 64 + lane];
#pragma unroll
            for (int kk = 0; kk < IN_F; kk += 4) {
                // A 16x4 fragment: contiguous float2 within the gathered row.
                v2f a  = *(const v2f*)(xrow + kk + ko);
                // B 4x16 fragments from conflict-free LDS stream.
                v2f b0 = wst[(kk >> 2) * 64];
                v2f b1 = wst[(kk >> 2) * 64 + 32];
                // 8-arg form: (neg_a, A, neg_b, B, c_mod, C, reuse_a, reuse_b)
                c0 = __builtin_amdgcn_wmma_f32_16x16x4_f32(false, a, false, b0,
                                                           (short)0, c0, false, false);
                c1 = __builtin_amdgcn_wmma_f32_16x16x4_f32(false, a, false, b1,
                                                           (short)0, c1, false, false);
            }
        }

        // Epilogue: bias + store per documented 16x16 f32 C/D layout:
        //   lanes 0-15: VGPR r holds (M=r,   N=lane)
        //   lanes16-31: VGPR r holds (M=r+8, N=lane-16)
        const float bv0 = bias[m];
        const float bv1 = bias[m + 16];
        const int rAdd = (lane < 16) ? 0 : 8;
#pragma unroll
        for (int r = 0; r < 8; ++r) {
            const int row = rowBase + r + rAdd;
            if (row < nNodes) {
                out[(size_t)row * OUT_F + m]      = c0[r] + bv0;
                out[(size_t)row * OUT_F + m + 16] = c1[r] + bv1;
            }
        }
    }
}

extern "C" void kernel_launch(void* const* d_in, const int* in_sizes, int n_in,
                              void* d_out, int out_size, void* d_ws, size_t ws_size,
                              hipStream_t stream)
{
    const float*     x     = (const float*)d_in[0];
    const long long* neigh = (const long long*)d_in[1];   // int64 indices
    const float*     W     = (const float*)d_in[2];
    const float*     b     = (const float*)d_in[3];
    float*           out   = (float*)d_out;

    const int nNodes = in_sizes[0] / IN_F;                // 163842
    const int nTiles = (nNodes + 15) / 16;                // 10241
    const int wavesPerBlock = 8;                          // 256 threads
    const int blocks = (nTiles + wavesPerBlock - 1) / wavesPerBlock;

    onering_conv_wmma_f32<<<blocks, wavesPerBlock * 32, 0, stream>>>(
        x, neigh, W, b, out, nNodes, nTiles);
}